// EntropyBottleneck_77747497992520
// MI455X (gfx1250) — compile-verified
//
#include <hip/hip_runtime.h>
#include <math.h>

// Problem constants (match reference)
#define C_CH   192
#define HW_    16384            // 128*128
#define BATCH  8
#define NPC    (BATCH * HW_)    // elements per channel = 131072
#define NTOT   ((size_t)C_CH * NPC)  // 25,165,824
#define PSTRIDE 64              // floats per channel in param block (47 used)

typedef float v2f __attribute__((ext_vector_type(2)));

// ---------- packed helpers (v_pk_fma_f32 / v_pk_mul/add_f32) ----------
__device__ __forceinline__ v2f splat(float s) { return (v2f){s, s}; }
__device__ __forceinline__ v2f pfma(v2f a, v2f b, v2f c) {
    return __builtin_elementwise_fma(a, b, c);
}

// ---------- trans ops (probe-confirmed: builtins exist on gfx1250) ----------
__device__ __forceinline__ v2f tanh2(v2f v) {
#if __has_builtin(__builtin_amdgcn_tanhf)
    v2f r; r.x = __builtin_amdgcn_tanhf(v.x); r.y = __builtin_amdgcn_tanhf(v.y);
    return r;
#else
    float rx, ry;
    asm("v_tanh_f32 %0, %2\n\t"
        "v_tanh_f32 %1, %3\n\t"
        "v_nop"
        : "=&v"(rx), "=&v"(ry) : "v"(v.x), "v"(v.y));
    return (v2f){rx, ry};
#endif
}
__device__ __forceinline__ v2f exp2_2(v2f v) {
#if __has_builtin(__builtin_amdgcn_exp2f)
    v2f r; r.x = __builtin_amdgcn_exp2f(v.x); r.y = __builtin_amdgcn_exp2f(v.y);
    return r;
#else
    float rx, ry;
    asm("v_exp_f32 %0, %2\n\t"
        "v_exp_f32 %1, %3\n\t"
        "v_nop"
        : "=&v"(rx), "=&v"(ry) : "v"(v.x), "v"(v.y));
    return (v2f){rx, ry};
#endif
}
__device__ __forceinline__ v2f rcp2(v2f v) {
#if __has_builtin(__builtin_amdgcn_rcpf)
    v2f r; r.x = __builtin_amdgcn_rcpf(v.x); r.y = __builtin_amdgcn_rcpf(v.y);
    return r;
#else
    float rx, ry;
    asm("v_rcp_f32 %0, %2\n\t"
        "v_rcp_f32 %1, %3\n\t"
        "v_nop"
        : "=&v"(rx), "=&v"(ry) : "v"(v.x), "v"(v.y));
    return (v2f){rx, ry};
#endif
}
__device__ __forceinline__ float fast_tanh1(float x) {   // prep-kernel use
#if __has_builtin(__builtin_amdgcn_tanhf)
    return __builtin_amdgcn_tanhf(x);
#else
    float r;
    asm("v_tanh_f32 %0, %1\n\tv_nop" : "=&v"(r) : "v"(x));
    return r;
#endif
}
__device__ __forceinline__ v2f sigmoid2(v2f v) {
    v2f t = v * splat(-1.44269504088896340736f);   // pk_mul
    v2f e = exp2_2(t);
    return rcp2(e + splat(1.0f));                  // pk_add + rcp pair
}

// ---------- per-channel param layout (stride PSTRIDE floats) ----------
// [0..2]  w0            [3..5]  bL0 = b0-0.5*w0   [6..8]  bU0 = b0+0.5*w0
// [9..11] tanh(f0)
// [12..20] w1           [21..23] b1               [24..26] tanh(f1)
// [27..35] w2           [36..38] b2               [39..41] tanh(f2)
// [42..44] w3           [45] b3                   [46] median
__device__ __forceinline__ v2f logits2(v2f v, const float* __restrict__ p, int bo) {
    v2f a0 = pfma(splat(p[0]), v, splat(p[bo + 0]));
    v2f a1 = pfma(splat(p[1]), v, splat(p[bo + 1]));
    v2f a2 = pfma(splat(p[2]), v, splat(p[bo + 2]));
    a0 = pfma(splat(p[ 9]), tanh2(a0), a0);
    a1 = pfma(splat(p[10]), tanh2(a1), a1);
    a2 = pfma(splat(p[11]), tanh2(a2), a2);
    v2f c0 = pfma(splat(p[12]), a0, pfma(splat(p[13]), a1, pfma(splat(p[14]), a2, splat(p[21]))));
    v2f c1 = pfma(splat(p[15]), a0, pfma(splat(p[16]), a1, pfma(splat(p[17]), a2, splat(p[22]))));
    v2f c2 = pfma(splat(p[18]), a0, pfma(splat(p[19]), a1, pfma(splat(p[20]), a2, splat(p[23]))));
    c0 = pfma(splat(p[24]), tanh2(c0), c0);
    c1 = pfma(splat(p[25]), tanh2(c1), c1);
    c2 = pfma(splat(p[26]), tanh2(c2), c2);
    v2f d0 = pfma(splat(p[27]), c0, pfma(splat(p[28]), c1, pfma(splat(p[29]), c2, splat(p[36]))));
    v2f d1 = pfma(splat(p[30]), c0, pfma(splat(p[31]), c1, pfma(splat(p[32]), c2, splat(p[37]))));
    v2f d2 = pfma(splat(p[33]), c0, pfma(splat(p[34]), c1, pfma(splat(p[35]), c2, splat(p[38]))));
    d0 = pfma(splat(p[39]), tanh2(d0), d0);
    d1 = pfma(splat(p[40]), tanh2(d1), d1);
    d2 = pfma(splat(p[41]), tanh2(d2), d2);
    return pfma(splat(p[42]), d0, pfma(splat(p[43]), d1, pfma(splat(p[44]), d2, splat(p[45]))));
}

__device__ __forceinline__ void eb_pair(v2f xv, const float* __restrict__ p, float med,
                                        v2f& o_out, v2f& l_out) {
    v2f o;
    o.x = rintf(xv.x - med);                  // round-half-even, matches jnp.round
    o.y = rintf(xv.y - med);
    o = o + splat(med);                        // pk_add
    v2f L = logits2(o, p, 3);
    v2f U = logits2(o, p, 6);
    v2f su = L + U;                            // pk_add
    v2f s;                                     // s = -sign(L+U), sign(0)=0
    s.x = (su.x > 0.0f) ? -1.0f : ((su.x < 0.0f) ? 1.0f : 0.0f);
    s.y = (su.y > 0.0f) ? -1.0f : ((su.y < 0.0f) ? 1.0f : 0.0f);
    v2f d = sigmoid2(s * U) - sigmoid2(s * L); // pk_mul x2, pk_sub
    l_out.x = fmaxf(fabsf(d.x), 1e-9f);        // v_max_num_f32 with |src| modifier
    l_out.y = fmaxf(fabsf(d.y), 1e-9f);
    o_out = o;
}

// ---------- hot kernel: streaming elementwise -------------------------------
// grid = (NPC/4096, C) ; block = 256 ; each thread: 16 elements (2 iterations
// of 2 x float4), all within one (b,c) plane (4096 | 16384).
__global__ void __launch_bounds__(256)
eb_main(const float* __restrict__ x, const float* __restrict__ Pg,
        float* __restrict__ out, float* __restrict__ lik) {
    const int c = blockIdx.y;
    const float* __restrict__ pg = Pg + c * PSTRIDE;
    float p[47];
    #pragma unroll
    for (int i = 0; i < 47; ++i) p[i] = pg[i];   // uniform -> s_load / SGPRs
    const float med = p[46];

    const int chunk = blockIdx.x * 4096;          // element offset within channel
    const int b     = chunk >> 14;                // plane index (4096 | 16384)
    const size_t plane = ((size_t)(b * C_CH + c)) << 14;
    const int rbase = (chunk & (HW_ - 1)) + threadIdx.x * 4;

    #pragma unroll
    for (int k = 0; k < 2; ++k) {
        const size_t base0 = plane + (size_t)(rbase + k * 2048);
        const size_t base1 = base0 + 1024;

        const float4 xv0 = *(const float4*)(x + base0);
        const float4 xv1 = *(const float4*)(x + base1);

        v2f o00, l00, o01, l01, o10, l10, o11, l11;
        eb_pair((v2f){xv0.x, xv0.y}, p, med, o00, l00);
        eb_pair((v2f){xv0.z, xv0.w}, p, med, o01, l01);
        eb_pair((v2f){xv1.x, xv1.y}, p, med, o10, l10);
        eb_pair((v2f){xv1.z, xv1.w}, p, med, o11, l11);

        float4 ov0 = {o00.x, o00.y, o01.x, o01.y};
        float4 ov1 = {o10.x, o10.y, o11.x, o11.y};
        float4 lv0 = {l00.x, l00.y, l01.x, l01.y};
        float4 lv1 = {l10.x, l10.y, l11.x, l11.y};

        *(float4*)(out + base0) = ov0;
        *(float4*)(out + base1) = ov1;
        *(float4*)(lik + base0) = lv0;
        *(float4*)(lik + base1) = lv1;
    }
}

// ---------- prep kernel: per-channel parameters (tiny; loops kept rolled) ----
__global__ void eb_prep_params(const float* __restrict__ m0, const float* __restrict__ b0, const float* __restrict__ f0,
                               const float* __restrict__ m1, const float* __restrict__ b1, const float* __restrict__ f1,
                               const float* __restrict__ m2, const float* __restrict__ b2, const float* __restrict__ f2,
                               const float* __restrict__ m3, const float* __restrict__ b3, const float* __restrict__ q,
                               float* __restrict__ P) {
    int c = blockIdx.x * blockDim.x + threadIdx.x;
    if (c >= C_CH) return;
    float* p = P + c * PSTRIDE;
    // stable softplus: log1p(exp(x)) = max(x,0) + log1p(exp(-|x|))
    #define SP_(v) (fmaxf((v), 0.0f) + log1pf(expf(-fabsf((v)))))
    #pragma unroll 1
    for (int j = 0; j < 3; ++j) {
        float w = SP_(m0[c * 3 + j]);
        float bb = b0[c * 3 + j];
        p[0 + j]  = w;
        p[3 + j]  = bb - 0.5f * w;             // bias for lower pass
        p[6 + j]  = bb + 0.5f * w;             // bias for upper pass
        p[9 + j]  = fast_tanh1(f0[c * 3 + j]);
        p[21 + j] = b1[c * 3 + j];
        p[24 + j] = fast_tanh1(f1[c * 3 + j]);
        p[36 + j] = b2[c * 3 + j];
        p[39 + j] = fast_tanh1(f2[c * 3 + j]);
        p[42 + j] = SP_(m3[c * 3 + j]);
    }
    #pragma unroll 1
    for (int k = 0; k < 9; ++k) {
        p[12 + k] = SP_(m1[c * 9 + k]);
        p[27 + k] = SP_(m2[c * 9 + k]);
    }
    p[45] = b3[c];
    p[46] = q[c * 3 + 1];   // median = quantiles[c, 0, 1]
    #undef SP_
}

// ---------- launcher ----------
extern "C" void kernel_launch(void* const* d_in, const int* in_sizes, int n_in,
                              void* d_out, int out_size, void* d_ws, size_t ws_size,
                              hipStream_t stream) {
    const float* x  = (const float*)d_in[0];
    const float* m0 = (const float*)d_in[1];
    const float* b0 = (const float*)d_in[2];
    const float* f0 = (const float*)d_in[3];
    const float* m1 = (const float*)d_in[4];
    const float* b1 = (const float*)d_in[5];
    const float* f1 = (const float*)d_in[6];
    const float* m2 = (const float*)d_in[7];
    const float* b2 = (const float*)d_in[8];
    const float* f2 = (const float*)d_in[9];
    const float* m3 = (const float*)d_in[10];
    const float* b3 = (const float*)d_in[11];
    const float* q  = (const float*)d_in[12];

    float* out = (float*)d_out;            // first output: quantized values
    float* lik = (float*)d_out + NTOT;     // second output: likelihoods
    float* P   = (float*)d_ws;             // 192*64*4 = 48 KB param block

    eb_prep_params<<<1, 192, 0, stream>>>(m0, b0, f0, m1, b1, f1,
                                          m2, b2, f2, m3, b3, q, P);

    dim3 grid(NPC / 4096, C_CH);           // (32, 192)
    eb_main<<<grid, 256, 0, stream>>>(x, P, out, lik);
}